// ParallelHyenaFilter_20486994002658
// MI455X (gfx1250) — compile-verified
//
#include <hip/hip_runtime.h>
#include <hip/hip_bf16.h>
#include <math.h>

typedef __attribute__((ext_vector_type(2))) float v2f;
typedef __attribute__((ext_vector_type(8))) float v8f;

#define SEQ_LEN 8192
#define ORDER   64
#define D_MODEL 4096

// ---------------------------------------------------------------------------
// Kernel A: the tiny MLP   z(8192x3) -> h3(8192x64)
// 64 threads per sequence row, 4 rows per 256-thread block.
// w2/w3 (16KB each) staged in LDS once per block.
// ---------------------------------------------------------------------------
__global__ __launch_bounds__(256) void hyena_mlp_kernel(
    const float* __restrict__ z,  const float* __restrict__ w1, const float* __restrict__ b1,
    const float* __restrict__ w2, const float* __restrict__ b2,
    const float* __restrict__ w3, const float* __restrict__ b3,
    const float* __restrict__ freq, float* __restrict__ h3out)
{
    __shared__ float sw2[ORDER * ORDER];
    __shared__ float sw3[ORDER * ORDER];
    __shared__ float sh1[4][ORDER];
    __shared__ float sh2[4][ORDER];

    const int tid = threadIdx.x;
    for (int i = tid; i < ORDER * ORDER; i += 256) {
        sw2[i] = w2[i];
        sw3[i] = w3[i];
    }

    const int r = tid >> 6;     // row-in-block 0..3
    const int o = tid & 63;     // channel 0..63
    const int l = blockIdx.x * 4 + r;

    const float fo = freq[o];

    // stage 1: (L,3) @ (3,64)
    const float* zr = z + l * 3;
    float acc = b1[o]
              + zr[0] * w1[o * 3 + 0]
              + zr[1] * w1[o * 3 + 1]
              + zr[2] * w1[o * 3 + 2];
    sh1[r][o] = sinf(fo * acc);
    __syncthreads();            // covers sw2/sw3 fill + sh1 publish

    // stage 2: (L,64) @ (64,64)
    acc = b2[o];
    #pragma unroll 8
    for (int j = 0; j < ORDER; ++j) acc += sh1[r][j] * sw2[o * ORDER + j];
    sh2[r][o] = sinf(fo * acc);
    __syncthreads();

    // stage 3: (L,64) @ (64,64)
    acc = b3[o];
    #pragma unroll 8
    for (int j = 0; j < ORDER; ++j) acc += sh2[r][j] * sw3[o * ORDER + j];
    h3out[l * ORDER + o] = sinf(fo * acc);
}

// ---------------------------------------------------------------------------
// Kernel B: out[d][l] = exp(-t[l]*|mod_w[d]|) * sum_k h3[l][k] * wf[d][k]
// fp32 WMMA 16x16x4. M = d (A = wf tile), N = l (B = h3 tile), K = 64.
// One wave: one 16-row d-tile, 8 consecutive 16-col l-tiles (A reused 8x).
//
// Per-lane fragment layouts (CDNA5 ISA 7.12.2, wave32):
//   A 16x4 f32 : lanes 0-15 -> M=lane, VGPR0/1 = K0,K1 ; lanes 16-31 -> K2,K3
//   B 4x16 f32 : lanes 0-15 -> N=lane, VGPR0/1 = K0,K1 ; lanes 16-31 -> K2,K3
//   C 16x16    : lane = N column (lane&15), VGPR r = row M=r (+8 upper half)
// ---------------------------------------------------------------------------
__global__ __launch_bounds__(256) void hyena_gemm_decay_kernel(
    const float* __restrict__ h3, const float* __restrict__ wf,
    const float* __restrict__ t,  const float* __restrict__ mod_w,
    float* __restrict__ out)
{
    const int lane  = threadIdx.x & 31;
    const int warp  = threadIdx.x >> 5;
    const int dTile = blockIdx.x & 255;                 // 256 d-tiles share A within block
    const int lChunk = ((blockIdx.x >> 8) << 3) + warp; // 0..63
    const int d0    = dTile * 16;
    const int lBase = lChunk * 128;                     // 8 N-tiles of 16

    const int nrow = lane & 15;           // M index for A loads / N index for B & C
    const int kh   = (lane >> 4) * 2;     // K sub-offset for this lane half
    const int mofs = (lane >> 4) * 8;     // C rows: upper lane half holds M+8

    // A fragments: 16 k-steps of wf[d0+nrow][4s + kh .. +1]  (one b64 per step)
    v2f afrag[16];
    const float* wrow = wf + (d0 + nrow) * ORDER;
    #pragma unroll
    for (int s = 0; s < 16; ++s)
        afrag[s] = *(const v2f*)(wrow + 4 * s + kh);

    // per-wave decay coefficients for the 8 d-rows this lane owns
    float absmw[8];
    #pragma unroll
    for (int rr = 0; rr < 8; ++rr)
        absmw[rr] = fabsf(mod_w[d0 + mofs + rr]);

    #pragma unroll
    for (int nt = 0; nt < 8; ++nt) {
        const int l0 = lBase + nt * 16;
        const float* brow = h3 + (l0 + nrow) * ORDER + kh;

        v8f c = {};
        #pragma unroll
        for (int s = 0; s < 16; ++s) {
            v2f b = *(const v2f*)(brow + 4 * s);
            c = __builtin_amdgcn_wmma_f32_16x16x4_f32(
                    /*neg_a=*/false, afrag[s],
                    /*neg_b=*/false, b,
                    /*c_mod=*/(short)0, c,
                    /*reuse_a=*/false, /*reuse_b=*/false);
        }

        const int lcol = l0 + nrow;       // this lane's sequence position
        const float tv = t[lcol];
        #pragma unroll
        for (int rr = 0; rr < 8; ++rr) {
            const float dec = __expf(-tv * absmw[rr]);
            // 134MB streaming output: keep it out of L2 with NT stores
            __builtin_nontemporal_store(c[rr] * dec,
                                        out + (size_t)(d0 + mofs + rr) * SEQ_LEN + lcol);
        }
    }
}

extern "C" void kernel_launch(void* const* d_in, const int* in_sizes, int n_in,
                              void* d_out, int out_size, void* d_ws, size_t ws_size,
                              hipStream_t stream) {
    // setup_inputs order: L, z, t, w1, b1, w2, b2, w3, b3, freq, wf, mod_w
    const float* z     = (const float*)d_in[1];
    const float* t     = (const float*)d_in[2];
    const float* w1    = (const float*)d_in[3];
    const float* b1    = (const float*)d_in[4];
    const float* w2    = (const float*)d_in[5];
    const float* b2    = (const float*)d_in[6];
    const float* w3    = (const float*)d_in[7];
    const float* b3    = (const float*)d_in[8];
    const float* freq  = (const float*)d_in[9];
    const float* wf    = (const float*)d_in[10];
    const float* mod_w = (const float*)d_in[11];
    float* out = (float*)d_out;
    float* h3  = (float*)d_ws;   // 8192*64 fp32 = 2 MB scratch

    hyena_mlp_kernel<<<SEQ_LEN / 4, 256, 0, stream>>>(
        z, w1, b1, w2, b2, w3, b3, freq, h3);

    // 256 d-tiles x 64 l-chunks, 8 waves (l-chunks) per block -> 2048 blocks
    hyena_gemm_decay_kernel<<<(D_MODEL / 16) * (SEQ_LEN / 128) / 8, 256, 0, stream>>>(
        h3, wf, t, mod_w, out);
}